// SideChainPacking_52458730553493
// MI455X (gfx1250) — compile-verified
//
#include <hip/hip_runtime.h>

typedef __attribute__((ext_vector_type(2))) float v2f;
typedef __attribute__((ext_vector_type(8))) float v8f;

#define NPTS   16384
#define CUT2   64.0f
#define JCHUNK 2048
#define NTILES (JCHUNK / 16)
#define BLOCK  256
#define WPB    (BLOCK / 32)        // 8 waves per block (wave32)
#define NJ     (NPTS / JCHUNK)     // 8 j-chunks
#define NIGRP  (NPTS / 16 / WPB)   // 128 i-tile groups

__global__ void SideChain_zero_kernel(float* out) {
    out[0] = 0.0f;
}

__global__ __launch_bounds__(BLOCK)
void SideChain_neighbor_wmma_kernel(const float* __restrict__ coords,
                                    const int*   __restrict__ seq_ids,
                                    const float* __restrict__ hydro_table,
                                    float*       __restrict__ out) {
    // Pre-swizzled WMMA B operands: per 16-point tile, 32 v2f entries laid out
    // exactly as V_WMMA_F32_16X16X4_F32 wants its B matrix (4x16, K split
    // across lane halves):
    //   sB[tile*32 + n]      = (x_n, y_n)   -> lanes 0-15  (K0, K1)
    //   sB[tile*32 + 16 + n] = (z_n, 1.0)   -> lanes 16-31 (K2, K3)
    __shared__ v2f   sB[JCHUNK * 2];   // 32 KB
    __shared__ float sT[JCHUNK];       // 8 KB : thr_j = 64 - |p_j|^2
    __shared__ float sPart[WPB];       // per-wave partials

    const int tid  = threadIdx.x;
    const int lane = tid & 31;
    const int wave = tid >> 5;
    const int l16  = lane & 15;
    const bool hi  = (lane >= 16);

    // Each wave owns one 16-row i-tile; each block owns one j-chunk.
    const int wid   = blockIdx.x * WPB + wave;
    const int i0    = wid * 16;
    const int jbase = blockIdx.y * JCHUNK;

    // ---- Build A (16x4 f32 WMMA layout) once per wave ----
    // A[m][0..2] = -2*coord_i[m],  A[m][3] = |coord_i[m]|^2
    // lanes 0-15 : v0 = K0 = -2x, v1 = K1 = -2y   (M = lane)
    // lanes 16-31: v0 = K2 = -2z, v1 = K3 = sq_i  (M = lane-16)
    const int im = i0 + l16;
    const float xi = coords[3 * im + 0];
    const float yi = coords[3 * im + 1];
    const float zi = coords[3 * im + 2];
    const float sqi = xi * xi + yi * yi + zi * zi;

    v2f A;
    A.x = hi ? (-2.0f * zi) : (-2.0f * xi);
    A.y = hi ? sqi          : (-2.0f * yi);

    // Cooperative stage of this block's j-chunk (B operands pre-swizzled).
    for (int t = tid; t < JCHUNK; t += BLOCK) {
        const int j = jbase + t;
        const float x = coords[3 * j + 0];
        const float y = coords[3 * j + 1];
        const float z = coords[3 * j + 2];
        const float sq = x * x + y * y + z * z;
        const int tile = t >> 4;
        const int pos  = t & 15;
        v2f blo; blo.x = x; blo.y = y;
        v2f bhi; bhi.x = z; bhi.y = 1.0f;
        sB[tile * 32 + pos]      = blo;
        sB[tile * 32 + 16 + pos] = bhi;
        sT[t] = CUT2 - sq;
    }
    __syncthreads();

    // Per-lane neighbor counters: cnt[r] covers row m = r + 8*hi,
    // column n = lane&15 of each j-tile, summed over this j-chunk.
    unsigned cnt[8];
    #pragma unroll
    for (int r = 0; r < 8; ++r) cnt[r] = 0u;

    #pragma unroll 4
    for (int tt = 0; tt < NTILES; ++tt) {
        const v2f   B   = sB[tt * 32 + lane];   // ds_load_b64, layout-ready
        const float thr = sT[tt * 16 + l16];    // ds_load_b32 (broadcast)

        v8f Cz = {0.0f, 0.0f, 0.0f, 0.0f, 0.0f, 0.0f, 0.0f, 0.0f};
        // D[m][n] = sq_i[m] - 2 * dot(coord_i[m], coord_j[n])
        v8f D = __builtin_amdgcn_wmma_f32_16x16x4_f32(
            false, A, false, B, (short)0, Cz, false, false);

        // d2 < 64  <=>  D < thr ; count via cmp + add-with-carry
        #pragma unroll
        for (int r = 0; r < 8; ++r) {
            cnt[r] += (D[r] < thr) ? 1u : 0u;
        }
    }

    // ---- Weight rows by hydrophobicity, reduce wave -> block -> global ----
    float p = 0.0f;
    #pragma unroll
    for (int r = 0; r < 8; ++r) {
        const int row = i0 + r + (hi ? 8 : 0);   // m for (r, half-wave)
        const float h = hydro_table[seq_ids[row]];
        p += h * (float)cnt[r];
    }

    #pragma unroll
    for (int off = 16; off > 0; off >>= 1) {
        p += __shfl_xor(p, off, 32);
    }
    if (lane == 0) sPart[wave] = p;
    __syncthreads();

    if (tid == 0) {
        float s = 0.0f;
        #pragma unroll
        for (int w = 0; w < WPB; ++w) s += sPart[w];
        atomicAdd(out, s * (-0.1f / (float)NPTS));
    }
}

extern "C" void kernel_launch(void* const* d_in, const int* in_sizes, int n_in,
                              void* d_out, int out_size, void* d_ws, size_t ws_size,
                              hipStream_t stream) {
    const float* coords      = (const float*)d_in[0];   // [N,3] f32
    const int*   seq_ids     = (const int*)d_in[1];     // [N]   i32
    const float* hydro_table = (const float*)d_in[2];   // [21]  f32
    float*       out         = (float*)d_out;           // [1]   f32

    SideChain_zero_kernel<<<dim3(1), dim3(1), 0, stream>>>(out);

    // 128 i-groups x 8 j-chunks = 1024 blocks = 8192 waves: enough to fill
    // every SIMD32 on the part several times over while staying L2-resident.
    SideChain_neighbor_wmma_kernel<<<dim3(NIGRP, NJ), dim3(BLOCK), 0, stream>>>(
        coords, seq_ids, hydro_table, out);
}